// AdaptivePatcher_84473416777772
// MI455X (gfx1250) — compile-verified
//
#include <hip/hip_runtime.h>
#include <hip/hip_bf16.h>
#include <stdint.h>

#define B      32
#define S      16384
#define MAXPS  8
#define CHUNK  1024
#define WPS    (S / 32)        /* 512 hi-bit words per sequence */

#ifndef __has_builtin
#define __has_builtin(x) 0
#endif

#if __has_builtin(__builtin_amdgcn_global_load_async_to_lds_b128)
#define HAVE_ASYNC_LDS 1
#else
#define HAVE_ASYNC_LDS 0
#endif

typedef int v4i __attribute__((vector_size(16)));
typedef __attribute__((address_space(1))) v4i* gv4i_p;   // global AS pointer
typedef __attribute__((address_space(3))) v4i* lv4i_p;   // LDS AS pointer

__device__ __forceinline__ void wait_async_zero() {
#if __has_builtin(__builtin_amdgcn_s_wait_asynccnt)
    __builtin_amdgcn_s_wait_asynccnt(0);
#else
    asm volatile("s_wait_asynccnt 0" ::: "memory");
#endif
}

// ---------------------------------------------------------------------------
// K1: hi[p] = (entropy of window >= 1.2 bits)  ==  >=3 distinct tokens.
// Stage CHUNK(+8) tokens to LDS with async global->LDS, pack bits via ballot.
// ---------------------------------------------------------------------------
__global__ void hi_kernel(const int* __restrict__ tokens,
                          unsigned* __restrict__ hi_bits) {
    __shared__ __align__(16) int sm[CHUNK + 8];
    const int blk  = blockIdx.x;             // B * (S/CHUNK) = 512 blocks
    const int b    = blk / (S / CHUNK);
    const int c    = (blk % (S / CHUNK)) * CHUNK;
    const int tid  = threadIdx.x;            // 256 threads
    const int* g   = tokens + (size_t)b * S + c;

#if HAVE_ASYNC_LDS
    // 256 lanes x 16B = 1024 ints; 2 lanes fetch the 8-int lookahead tail.
    __builtin_amdgcn_global_load_async_to_lds_b128(
        (gv4i_p)(g + tid * 4), (lv4i_p)(&sm[tid * 4]), 0, 0);
    if (tid < 2 && c + CHUNK < S) {
        __builtin_amdgcn_global_load_async_to_lds_b128(
            (gv4i_p)(g + CHUNK + tid * 4), (lv4i_p)(&sm[CHUNK + tid * 4]), 0, 0);
    }
    wait_async_zero();
#else
    for (int i = tid; i < CHUNK; i += 256) sm[i] = g[i];
    if (tid < 8 && c + CHUNK < S) sm[CHUNK + tid] = g[CHUNK + tid];
#endif
    __syncthreads();

    const int warp = tid >> 5;
    const int lane = tid & 31;
    for (int it = 0; it < 4; ++it) {
        const int li = (warp * 4 + it) * 32 + lane;   // 0..1023
        const int p  = c + li;
        const int t0 = sm[li], t1 = sm[li + 1], t2 = sm[li + 2], t3 = sm[li + 3];
        bool hi;
        if (p + 3 < S) {
            // equal-pair count: {4}=6 {3,1}=3 {2,2}=2 {2,1,1}=1 {1,1,1,1}=0
            int e = (t0 == t1) + (t0 == t2) + (t0 == t3)
                  + (t1 == t2) + (t1 == t3) + (t2 == t3);
            hi = (e <= 1);                    // >=3 distinct -> H >= 1.5
        } else if (p + 3 == S) {              // 3-token tail window
            hi = (t0 != t1) && (t0 != t2) && (t1 != t2);  // H = 1.585
        } else {                              // 1- or 2-token window: H <= 1.0
            hi = false;
        }
        unsigned m = (unsigned)__ballot(hi);  // wave32: low 32 bits
        if (lane == 0)
            hi_bits[(size_t)b * WPS + (c >> 5) + warp * 4 + it] = m;
    }
}

// ---------------------------------------------------------------------------
// K2: sequential patch walk, one lane per sequence; all bitmasks in LDS.
// size(i) = clamp(next_hi[i+1] - i, 1, min(8, S - i))
// ---------------------------------------------------------------------------
__global__ void walk_kernel(const unsigned* __restrict__ hi_bits,
                            int* __restrict__ offs,
                            unsigned char* __restrict__ sizes,
                            int* __restrict__ counts) {
    __shared__ unsigned bits[B * WPS];       // 64 KB
    const int tid = threadIdx.x;             // 256 threads
    for (int k = tid; k < B * WPS; k += 256) bits[k] = hi_bits[k];
    __syncthreads();

    if (tid < B) {
        const int b = tid;
        const unsigned* wb = &bits[b * WPS];
        int i = 0, n = 0;
        while (i < S) {
            const int rem   = S - i;
            const int limit = rem < MAXPS ? rem : MAXPS;
            int width = S - 1 - i;               // # candidate positions q=i+1..
            if (width > 8) width = 8;
            unsigned m = 0u;
            if (width > 0) {
                const int q0 = i + 1;
                const int w  = q0 >> 5, sh = q0 & 31;
                const unsigned lo  = wb[w];
                const unsigned hiw = (w + 1 < WPS) ? wb[w + 1] : 0u;
                uint64_t cc = (((uint64_t)hiw << 32) | lo) >> sh;
                const unsigned wm = (width >= 8) ? 0xFFu : ((1u << width) - 1u);
                m = (unsigned)cc & wm;
            }
            const int d    = m ? (__builtin_ctz(m) + 1) : 9;  // next_hi - i (capped)
            const int size = d < limit ? d : limit;
            offs[(size_t)b * S + n]  = i;
            sizes[(size_t)b * S + n] = (unsigned char)size;
            ++n;
            i += size;
        }
        counts[b] = n;
    }
}

// ---------------------------------------------------------------------------
// K3: write padded patches [B,S,8] and offsets [B,S] (float; -1 past end).
// ---------------------------------------------------------------------------
__global__ void fill_kernel(const int* __restrict__ tokens,
                            const int* __restrict__ offs,
                            const unsigned char* __restrict__ sizes,
                            const int* __restrict__ counts,
                            float* __restrict__ patches_out,
                            float* __restrict__ offsets_out) {
    const int idx = blockIdx.x * blockDim.x + threadIdx.x;   // over B*S
    if (idx >= B * S) return;
    const int b = idx / S;
    const int p = idx - b * S;
    const int n = counts[b];
    int off = 0, sz = 0;
    float offv = -1.0f;
    if (p < n) {
        off  = offs[idx];
        sz   = sizes[idx];
        offv = (float)off;
    }
    offsets_out[idx] = offv;
    const int* seq = tokens + (size_t)b * S;
    float* dst = patches_out + (size_t)idx * MAXPS;
#pragma unroll
    for (int j = 0; j < MAXPS; ++j)
        dst[j] = (j < sz) ? (float)seq[off + j] : 0.0f;       // PAD = 0
}

// ---------------------------------------------------------------------------
// K4: num_patches [B] and cu_seqlens [B+1].
// ---------------------------------------------------------------------------
__global__ void tail_kernel(const int* __restrict__ counts,
                            float* __restrict__ n_out,
                            float* __restrict__ cu_out) {
    if (threadIdx.x == 0 && blockIdx.x == 0) {
        int acc = 0;
        cu_out[0] = 0.0f;
        for (int b = 0; b < B; ++b) {
            const int c = counts[b];
            n_out[b] = (float)c;
            acc += c;
            cu_out[b + 1] = (float)acc;
        }
    }
}

extern "C" void kernel_launch(void* const* d_in, const int* in_sizes, int n_in,
                              void* d_out, int out_size, void* d_ws, size_t ws_size,
                              hipStream_t stream) {
    (void)in_sizes; (void)n_in; (void)out_size; (void)ws_size;
    const int* tokens = (const int*)d_in[0];

    // d_out layout (flat float, reference return order):
    // patches [B*S*8] | offsets [B*S] | n [B] | cu_seqlens [B+1]
    float* out       = (float*)d_out;
    float* patches_o = out;
    float* offsets_o = out + (size_t)B * S * MAXPS;
    float* n_o       = offsets_o + (size_t)B * S;
    float* cu_o      = n_o + B;

    // workspace: hi_bits (64KB) | offs (2MB int) | sizes (512KB u8) | counts
    char* ws = (char*)d_ws;
    unsigned*      hi_bits = (unsigned*)ws;
    int*           offs    = (int*)(ws + (size_t)B * WPS * 4);
    unsigned char* sizes   = (unsigned char*)(ws + (size_t)B * WPS * 4 + (size_t)B * S * 4);
    int*           counts  = (int*)(ws + (size_t)B * WPS * 4 + (size_t)B * S * 4 + (size_t)B * S);

    hi_kernel  <<<B * (S / CHUNK), 256, 0, stream>>>(tokens, hi_bits);
    walk_kernel<<<1, 256, 0, stream>>>(hi_bits, offs, sizes, counts);
    fill_kernel<<<(B * S + 255) / 256, 256, 0, stream>>>(tokens, offs, sizes, counts,
                                                         patches_o, offsets_o);
    tail_kernel<<<1, 32, 0, stream>>>(counts, n_o, cu_o);
}